// FinePreprocess_51848845197542
// MI455X (gfx1250) — compile-verified
//
#include <hip/hip_runtime.h>

typedef __attribute__((ext_vector_type(2))) float v2f;
typedef __attribute__((ext_vector_type(8))) float v8f;

#define CCOARSE 256
#define CFINE   128
#define WIN     5
#define WW      25           // WIN*WIN
#define MPB     2            // matches per block
#define ROWS    32           // padded rows per match (25 -> 32)
#define ASTR    132          // LDS row stride in floats (128+4, bank-conflict free)

// ---------------------------------------------------------------------------
// Kernel 1: fold W_dp into the second half of W_merge.
//   Wc[n][k] = sum_j W_merge[n][128+j] * W_dp[j][k]        (128 x 256)
//   bc[n]    = b_merge[n] + sum_j b_dp[j] * W_merge[n][128+j]
// ---------------------------------------------------------------------------
__global__ __launch_bounds__(256)
void wc_precompute(const float* __restrict__ W_merge,   // (128, 256)
                   const float* __restrict__ W_dp,      // (128, 256)
                   const float* __restrict__ b_dp,      // (128)
                   const float* __restrict__ b_merge,   // (128)
                   float* __restrict__ Wc,              // (128, 256)
                   float* __restrict__ bc) {            // (128)
  const int n = blockIdx.x;      // 0..127
  const int k = threadIdx.x;     // 0..255
  const float* wm2 = W_merge + (size_t)n * (2 * CFINE) + CFINE;
  float acc = 0.f;
  for (int j = 0; j < CFINE; ++j)
    acc += wm2[j] * W_dp[(size_t)j * CCOARSE + k];
  Wc[(size_t)n * CCOARSE + k] = acc;
  if (k == 0) {
    float a = b_merge[n];
    for (int j = 0; j < CFINE; ++j)
      a += b_dp[j] * wm2[j];
    bc[n] = a;
  }
}

// ---------------------------------------------------------------------------
// Kernel 2: gather windows + fused GEMM via V_WMMA_F32_16X16X4_F32.
// Block handles MPB matches; 16 waves: wave w -> N-tile (w%8), M-group (w/8).
// out[g, p, n] = sum_k F[g,p,k]*Wm1[n,k] + t[g,n],  g in [0, 2M)
// ---------------------------------------------------------------------------
__global__ __launch_bounds__(512)
void fine_preprocess(const float* __restrict__ cf1,
                     const float* __restrict__ cf2,
                     const float* __restrict__ ff1,
                     const float* __restrict__ ff2,
                     const int*   __restrict__ bidx,
                     const int*   __restrict__ ridx,
                     const int*   __restrict__ cidx,
                     const int*   __restrict__ pHf,
                     const int*   __restrict__ pHc,
                     const float* __restrict__ W_merge, // (128, 256); Wm1 = cols 0..127
                     const float* __restrict__ Wc,      // folded coarse weight
                     const float* __restrict__ bc,      // folded bias
                     float* __restrict__ out,           // (2M, 25, 128)
                     int M) {
  __shared__ float ldsA[MPB * ROWS * ASTR];   // window tiles, zero padded rows
  __shared__ float ldsC[MPB * CCOARSE];       // coarse rows
  __shared__ float ldsT[MPB * CFINE];         // folded coarse terms

  const int tid    = threadIdx.x;
  const int Hf     = pHf[0];
  const int Hc     = pHc[0];
  const int stride = Hf / Hc;
  const int pad    = WIN / 2;
  const int hw2    = Hc * Hc;
  const int m0     = blockIdx.x * MPB;
  const int totalM = 2 * M;

  // ---- zero the padded rows 25..31 of each A tile ----
  for (int i = tid; i < MPB * (ROWS - WW) * CFINE; i += blockDim.x) {
    int lm = i / ((ROWS - WW) * CFINE);
    int r  = (i / CFINE) % (ROWS - WW) + WW;
    int ch = i % CFINE;
    ldsA[(lm * ROWS + r) * ASTR + ch] = 0.f;
  }

  // ---- stage coarse rows and gather fine windows into LDS ----
  for (int lm = 0; lm < MPB; ++lm) {
    const int g      = m0 + lm;
    const bool valid = (g < totalM);
    const bool first = (g < M);
    const int gm     = valid ? (first ? g : g - M) : 0;
    const float* fine   = first ? ff1 : ff2;
    const float* coarse = first ? cf1 : cf2;
    // indices are block-uniform: pin them to SGPRs
    const int b   = __builtin_amdgcn_readfirstlane(bidx[gm]);
    const int lin = __builtin_amdgcn_readfirstlane(first ? ridx[gm] : cidx[gm]);
    const int r   = lin / Hc;
    const int c   = lin % Hc;

    for (int k = tid; k < CCOARSE; k += blockDim.x)
      ldsC[lm * CCOARSE + k] =
          valid ? coarse[((size_t)b * hw2 + lin) * CCOARSE + k] : 0.f;

    for (int i = tid; i < CFINE * WW; i += blockDim.x) {
      const int ch  = i / WW;
      const int pos = i % WW;          // consecutive tids walk x (contiguous)
      const int wy  = pos / WIN;
      const int wx  = pos % WIN;
      const int y   = r * stride - pad + wy;
      const int x   = c * stride - pad + wx;
      float v = 0.f;
      if (valid && y >= 0 && y < Hf && x >= 0 && x < Hf)
        v = fine[(((size_t)b * CFINE + ch) * Hf + y) * Hf + x];
      ldsA[(lm * ROWS + pos) * ASTR + ch] = v;
    }
  }

  // ---- per-wave B preload: 16x128 block of Wm1 into registers ----
  // B(4x16) layout: VGPR0 = B[k0+khalf][n], VGPR1 = B[k0+khalf+1][n]
  // with khalf = 0 for lanes 0-15, 2 for lanes 16-31; n striped over lane%16.
  const int lane  = tid & 31;
  // wave id is uniform within the wave: make it an SGPR so the mt-loop below
  // becomes scalar control flow (EXEC stays all-1s around the WMMAs).
  const int wv    = __builtin_amdgcn_readfirstlane(tid >> 5);  // 0..15
  const int nt    = wv & 7;          // N tile (16 channels each)
  const int mg    = wv >> 3;         // M-group 0/1
  const int nn    = nt * 16 + (lane & 15);
  const int khalf = (lane < 16) ? 0 : 2;
  const float* wrow = W_merge + (size_t)nn * (2 * CFINE);  // Wm1[n][k] = W_merge[n][k]
  v2f bfrag[32];
#pragma unroll
  for (int ks = 0; ks < 32; ++ks)
    bfrag[ks] = *(const v2f*)(wrow + ks * 4 + khalf);

  __syncthreads();

  // ---- folded coarse term: t[lm][n] = bc[n] + coarse_row . Wc[n] ----
  if (tid < MPB * CFINE) {
    const int lm = tid / CFINE;
    const int n  = tid % CFINE;
    const float* wcrow = Wc + (size_t)n * CCOARSE;
    float acc = bc[n];
    for (int k = 0; k < CCOARSE; ++k)
      acc += ldsC[lm * CCOARSE + k] * wcrow[k];
    ldsT[tid] = acc;
  }
  __syncthreads();

  // ---- WMMA GEMM: per wave, loop over its M tiles (scalar control flow) ----
  for (int mt = mg; mt < MPB * 2; mt += 2) {
    const int lm = mt >> 1;
    const int rb = (mt & 1) * 16;
    // A(16x4) layout: lane holds row rb + lane%16; VGPR0 = K=k0+khalf, VGPR1 = +1
    const float* arow = &ldsA[(lm * ROWS + rb + (lane & 15)) * ASTR + khalf];

    v8f acc = {0.f, 0.f, 0.f, 0.f, 0.f, 0.f, 0.f, 0.f};
#pragma unroll
    for (int ks = 0; ks < 32; ++ks) {
      v2f a = *(const v2f*)(arow + ks * 4);        // ds_load_b64, conflict-free
      acc = __builtin_amdgcn_wmma_f32_16x16x4_f32(
          false, a, false, bfrag[ks], (short)0, acc, false, false);
    }

    const int g = m0 + lm;
    if (g < totalM) {
      const float tadd = ldsT[lm * CFINE + nn];
      const int hi = (lane < 16) ? 0 : 8;
      float* obase = out + (size_t)g * WW * CFINE + nn;
#pragma unroll
      for (int i = 0; i < 8; ++i) {
        const int wr = rb + i + hi;    // row within 25-row window (D layout)
        if (wr < WW)
          obase[(size_t)wr * CFINE] = acc[i] + tadd;
      }
    }
  }
}

// ---------------------------------------------------------------------------
extern "C" void kernel_launch(void* const* d_in, const int* in_sizes, int n_in,
                              void* d_out, int out_size, void* d_ws, size_t ws_size,
                              hipStream_t stream) {
  const float* cf1     = (const float*)d_in[0];
  const float* cf2     = (const float*)d_in[1];
  const float* ff1     = (const float*)d_in[2];
  const float* ff2     = (const float*)d_in[3];
  const int*   bidx    = (const int*)d_in[4];
  const int*   ridx    = (const int*)d_in[5];
  const int*   cidx    = (const int*)d_in[6];
  const int*   pHf     = (const int*)d_in[7];
  const int*   pHc     = (const int*)d_in[8];
  const float* W_dp    = (const float*)d_in[9];
  const float* b_dp    = (const float*)d_in[10];
  const float* W_merge = (const float*)d_in[11];
  const float* b_merge = (const float*)d_in[12];

  const int M = in_sizes[4];

  float* Wc = (float*)d_ws;              // 128*256 floats
  float* bc = Wc + CFINE * CCOARSE;      // 128 floats

  wc_precompute<<<CFINE, 256, 0, stream>>>(W_merge, W_dp, b_dp, b_merge, Wc, bc);

  const int blocks = (2 * M + MPB - 1) / MPB;
  fine_preprocess<<<blocks, 512, 0, stream>>>(
      cf1, cf2, ff1, ff2, bidx, ridx, cidx, pHf, pHc,
      W_merge, Wc, bc, (float*)d_out, M);
}